// CriticalField_39436389712434
// MI455X (gfx1250) — compile-verified
//
#include <hip/hip_runtime.h>
#include <cstdint>

// CriticalField: periodic 5-point stencil + pointwise update on two f32 fields.
// Bandwidth-bound (≈268 MB traffic, ~11.5 us floor at 23.3 TB/s on MI455X).
// CDNA5 paths: global_load_async_to_lds_b128 staging + s_wait_asynccnt,
// vec4 LDS reads (ds_load_b128) and vec4 stores (global_store_b128).

#define SIZE   4096
#define MASK   4095
#define TILE_H 8
#define TILE_W 512
#define HALO_ROWS (TILE_H + 2)              // 10 staged rows per block
#define VEC_PER_ROW (TILE_W / 4)            // 128 float4 per row
#define NTHREADS 256

__global__ __launch_bounds__(NTHREADS)
void CriticalField_39436389712434_kernel(const float* __restrict__ mag,
                                         const float* __restrict__ phs,
                                         const float* __restrict__ dampingP,
                                         const float* __restrict__ tensionP,
                                         const float* __restrict__ nonlinP,
                                         float* __restrict__ out)
{
    __shared__ __align__(16) float s_mag[HALO_ROWS * TILE_W];   // 20 KB
    __shared__ __align__(16) float s_phs[HALO_ROWS * TILE_W];   // 20 KB

    const float kCoupling = 0.015f;
    const float kDT       = 0.05f;
    const float kTwoPi    = 6.283185307179586f;

    const int tid    = threadIdx.x;
    const int tileX0 = blockIdx.x * TILE_W;
    const int tileY0 = blockIdx.y * TILE_H;

    // LDS byte addresses: flat pointer into LDS keeps the LDS offset in addr[31:0].
    const uint32_t lds_mag = (uint32_t)(uintptr_t)(&s_mag[0]);
    const uint32_t lds_phs = (uint32_t)(uintptr_t)(&s_phs[0]);

    // ---- Stage HALO_ROWS x TILE_W floats per field via async 128b copies ----
    // 10 rows * 128 vec4 = 1280 vec-loads per field; 5 loop iters/thread,
    // 2 async instructions per iter -> ASYNCcnt peaks at 10 per wave.
    for (int i = tid; i < HALO_ROWS * VEC_PER_ROW; i += NTHREADS) {
        const int r = i / VEC_PER_ROW;
        const int v = i - r * VEC_PER_ROW;
        const int grow = (tileY0 + r - 1) & MASK;           // wrap row (-1 -> 4095)
        const uint32_t goff = (uint32_t)(grow * SIZE + tileX0 + v * 4) * 4u;
        const uint32_t loff = (uint32_t)(r * TILE_W + v * 4) * 4u;
        const uint32_t lm = lds_mag + loff;
        const uint32_t lp = lds_phs + loff;
        asm volatile("global_load_async_to_lds_b128 %0, %1, %2 offset:0"
                     :: "v"(lm), "v"(goff), "s"(mag) : "memory");
        asm volatile("global_load_async_to_lds_b128 %0, %1, %2 offset:0"
                     :: "v"(lp), "v"(goff), "s"(phs) : "memory");
    }
    asm volatile("s_wait_asynccnt 0" ::: "memory");
    __syncthreads();

    const float damping = *dampingP;
    const float tension = *tensionP;
    const float nonlin  = *nonlinP;

    // ---- Compute: 4 vec4 tasks per thread per field, b128 LDS loads/stores ----
    const int VEC_PER_THREAD = (TILE_H * VEC_PER_ROW) / NTHREADS;  // 4
#pragma unroll
    for (int k = 0; k < VEC_PER_THREAD; ++k) {
        const int vj = tid + k * NTHREADS;   // 0..1023
        const int ly = vj >> 7;              // vj / 128
        const int vx = vj & (VEC_PER_ROW - 1);
        const int lx = vx * 4;
        const int gy = tileY0 + ly;
        const int gx = tileX0 + lx;

        const int c = (ly + 1) * TILE_W + lx;   // 16B-aligned element offset
        const float4 m  = *reinterpret_cast<const float4*>(&s_mag[c]);
        const float4 mu = *reinterpret_cast<const float4*>(&s_mag[c - TILE_W]);
        const float4 md = *reinterpret_cast<const float4*>(&s_mag[c + TILE_W]);
        const float4 p  = *reinterpret_cast<const float4*>(&s_phs[c]);
        const float4 pu = *reinterpret_cast<const float4*>(&s_phs[c - TILE_W]);
        const float4 pd = *reinterpret_cast<const float4*>(&s_phs[c + TILE_W]);

        // Shift-in scalars for the horizontal neighbors of the vec4.
        float ml, pl, mr, pr;
        if (vx > 0) {
            ml = s_mag[c - 1];
            pl = s_phs[c - 1];
        } else {                               // column wrap, hits L2
            const int gxl = (gx + MASK) & MASK;
            ml = mag[gy * SIZE + gxl];
            pl = phs[gy * SIZE + gxl];
        }
        if (vx < VEC_PER_ROW - 1) {
            mr = s_mag[c + 4];
            pr = s_phs[c + 4];
        } else {
            const int gxr = (gx + 4) & MASK;
            mr = mag[gy * SIZE + gxr];
            pr = phs[gy * SIZE + gxr];
        }

        // Horizontal neighbor vectors by register shifts (no extra LDS traffic).
        const float mlv[4] = { ml,  m.x, m.y, m.z };
        const float mrv[4] = { m.y, m.z, m.w, mr  };
        const float plv[4] = { pl,  p.x, p.y, p.z };
        const float prv[4] = { p.y, p.z, p.w, pr  };
        const float mc[4]  = { m.x,  m.y,  m.z,  m.w  };
        const float muv[4] = { mu.x, mu.y, mu.z, mu.w };
        const float mdv[4] = { md.x, md.y, md.z, md.w };
        const float pc[4]  = { p.x,  p.y,  p.z,  p.w  };
        const float puv[4] = { pu.x, pu.y, pu.z, pu.w };
        const float pdv[4] = { pd.x, pd.y, pd.z, pd.w };

        float nm[4], np[4];
#pragma unroll
        for (int e = 0; e < 4; ++e) {
            const float mlap = muv[e] + mdv[e] + mlv[e] + mrv[e] - 4.0f * mc[e];
            const float plap = puv[e] + pdv[e] + plv[e] + prv[e] - 4.0f * pc[e];
            const float dmag = tension * mlap - damping * mc[e]
                             - nonlin * (mc[e] * mc[e] * mc[e]);
            const float dphs = tension * plap + kCoupling * __sinf(puv[e] - pc[e]);
            nm[e] = fminf(fmaxf(mc[e] + kDT * dmag, -2.0f), 2.0f);
            np[e] = fminf(fmaxf(pc[e] + kDT * dphs, 0.0f), kTwoPi);
        }

        const float4 nm4 = make_float4(nm[0], nm[1], nm[2], nm[3]);
        const float4 np4 = make_float4(np[0], np[1], np[2], np[3]);
        *reinterpret_cast<float4*>(&out[gy * SIZE + gx]) = nm4;               // ch 0
        *reinterpret_cast<float4*>(&out[SIZE * SIZE + gy * SIZE + gx]) = np4; // ch 1
    }
}

extern "C" void kernel_launch(void* const* d_in, const int* in_sizes, int n_in,
                              void* d_out, int out_size, void* d_ws, size_t ws_size,
                              hipStream_t stream) {
    (void)in_sizes; (void)n_in; (void)out_size; (void)d_ws; (void)ws_size;
    const float* mag      = (const float*)d_in[0];
    const float* phs      = (const float*)d_in[1];
    const float* damping  = (const float*)d_in[2];
    const float* tension  = (const float*)d_in[3];
    const float* nonlin   = (const float*)d_in[4];
    float* out = (float*)d_out;

    dim3 grid(SIZE / TILE_W, SIZE / TILE_H);   // (8, 512) = 4096 blocks
    dim3 block(NTHREADS);
    CriticalField_39436389712434_kernel<<<grid, block, 0, stream>>>(
        mag, phs, damping, tension, nonlin, out);
}